// Qwen3VLMoeTextExperts_38843684225327
// MI455X (gfx1250) — compile-verified
//
#include <hip/hip_runtime.h>
#include <hip/hip_bf16.h>

// Problem constants (from reference)
#define E_    32
#define H_    2048
#define I_    768
#define T_    2048
#define K_    4
#define TWO_I 1536
#define NPAIR (T_ * K_)          // 8192 routed (token, expert) pairs
#define MAX_TILES 544            // sum ceil(cnt_e/16) <= 8192/16 + 32

typedef float v2f __attribute__((ext_vector_type(2)));
typedef float v8f __attribute__((ext_vector_type(8)));
typedef int   i32x2 __attribute__((ext_vector_type(2)));
typedef int   i32x4 __attribute__((ext_vector_type(4)));

// ---------------------------------------------------------------------------
// gfx1250 async global->LDS copies (ASYNCcnt-tracked, bypass VGPRs).
// Builtin params are (int-vector AS1*, int-vector AS3*, imm offset, imm cpol)
// per the compiler diagnostic. Guarded with a register-staging fallback.
// ---------------------------------------------------------------------------
#if defined(__has_builtin)
#if __has_builtin(__builtin_amdgcn_global_load_async_to_lds_b128) && \
    __has_builtin(__builtin_amdgcn_global_load_async_to_lds_b64)
#define USE_ASYNC_LDS 1
#endif
#endif
#ifndef USE_ASYNC_LDS
#define USE_ASYNC_LDS 0
#endif

#if USE_ASYNC_LDS
typedef __attribute__((address_space(1))) i32x4* gp128_t;
typedef __attribute__((address_space(3))) i32x4* lp128_t;
typedef __attribute__((address_space(1))) i32x2* gp64_t;
typedef __attribute__((address_space(3))) i32x2* lp64_t;

__device__ __forceinline__ void a_copy128(const float* g, float* l) {
    __builtin_amdgcn_global_load_async_to_lds_b128((gp128_t)g, (lp128_t)l, 0, 0);
}
__device__ __forceinline__ void a_copy64(const float* g, float* l) {
    __builtin_amdgcn_global_load_async_to_lds_b64((gp64_t)g, (lp64_t)l, 0, 0);
}
#endif

__device__ __forceinline__ void wait_async_zero() {
#if USE_ASYNC_LDS
#if __has_builtin(__builtin_amdgcn_s_wait_asynccnt)
    __builtin_amdgcn_s_wait_asynccnt(0);
#else
    asm volatile("s_wait_asynccnt 0" ::: "memory");
#endif
#endif
}

// ---------------------------------------------------------------------------
// Routing / bookkeeping kernels (tiny)
// ---------------------------------------------------------------------------
__global__ void k_zero_out(float4* __restrict__ out) {
    out[(size_t)blockIdx.x * blockDim.x + threadIdx.x] = float4{0.f, 0.f, 0.f, 0.f};
}

__global__ void k_route_zero(int* __restrict__ cnt) {
    if (threadIdx.x < E_) cnt[threadIdx.x] = 0;
}

__global__ void k_count(const int* __restrict__ ridx, int* __restrict__ cnt) {
    const int i = blockIdx.x * 256 + threadIdx.x;   // i < NPAIR
    atomicAdd(&cnt[ridx[i]], 1);
}

__global__ void k_scan(const int* __restrict__ cnt, int* __restrict__ off,
                       int* __restrict__ cursor, int* __restrict__ tile_e,
                       int* __restrict__ tile_row, int* __restrict__ tile_end,
                       int* __restrict__ numTiles) {
    if (threadIdx.x != 0) return;
    int acc = 0, idx = 0;
    for (int e = 0; e < E_; ++e) {
        off[e]    = acc;
        cursor[e] = acc;
        const int c   = cnt[e];
        const int end = acc + c;
        for (int r = acc; r < end; r += 16) {
            tile_e[idx]   = e;
            tile_row[idx] = r;
            tile_end[idx] = end;
            ++idx;
        }
        acc = end;
    }
    *numTiles = idx;
}

__global__ void k_scatter(const int* __restrict__ ridx, const float* __restrict__ rw,
                          int* __restrict__ cursor, int* __restrict__ pair_token,
                          float* __restrict__ pair_w) {
    const int i = blockIdx.x * 256 + threadIdx.x;   // i < NPAIR
    const int e = ridx[i];
    const int pos = atomicAdd(&cursor[e], 1);
    pair_token[pos] = i >> 2;                       // token id (K_ == 4)
    pair_w[pos]     = rw[i];
}

// ---------------------------------------------------------------------------
// Kernel: gate_up GEMM + SwiGLU.
// Block = (tile of 16 gathered tokens) x (128 columns of I).
// 8 wave32s; each wave owns one 16x16 gate tile + matching up tile.
// Double-buffered LDS; async staging overlapped with WMMA compute.
// ---------------------------------------------------------------------------
__global__ __launch_bounds__(256)
void k_gateup(const float* __restrict__ hs, const float* __restrict__ gup,
              const int* __restrict__ pair_token,
              const int* __restrict__ tile_e, const int* __restrict__ tile_row,
              const int* __restrict__ tile_end, const int* __restrict__ numTiles,
              float* __restrict__ gated) {
    const int tile = blockIdx.x;
    if (tile >= *numTiles) return;
    const int e        = tile_e[tile];
    const int rowStart = tile_row[tile];
    const int segEnd   = tile_end[tile];
    const int n0       = blockIdx.y * 128;   // column group within I

    const int tid  = threadIdx.x;
    const int lane = tid & 31;
    const int wv   = tid >> 5;               // wave 0..7
    const int cb   = wv * 16;                // wave's column base in the 128-group

    constexpr int AS = 36;                   // 32 + 4 pad -> conflict-free A frags
    constexpr int BS = 144;                  // 128 + 16 pad -> half-waves hit disjoint banks
    __shared__ float As[2][16 * AS];
    __shared__ float Bg[2][32 * BS];
    __shared__ float Bu[2][32 * BS];

    // A-staging assignment: 16 rows x 32 cols, one b64 per thread.
    const int arow = tid >> 4;               // 0..15
    const int acol = (tid & 15) * 2;         // 0..30
    int prow = rowStart + arow;
    if (prow >= segEnd) prow = segEnd - 1;   // clamp (masked at store)
    const float* hrow = hs + (size_t)pair_token[prow] * H_;

    // B-staging assignment: 32 rows x 128 cols, 4x b128 per thread per matrix.
    const int brow = tid >> 3;               // 0..31
    const int bcol = (tid & 7) * 16;         // 0,16,...,112
    const float* wbase = gup + (size_t)e * H_ * TWO_I;

    const int m    = lane & 15;              // A row / B column within tile
    const int half = lane >> 4;              // 0 or 1
    const int koff = half * 2;

    auto stage = [&](int kc, int buf) {
        const float* ga = hrow + kc + acol;
        float*       la = &As[buf][arow * AS + acol];
        const float* wg = wbase + (size_t)(kc + brow) * TWO_I + n0 + bcol;
        const float* wu = wg + I_;
        float*       lg = &Bg[buf][brow * BS + bcol];
        float*       lu = &Bu[buf][brow * BS + bcol];
#if USE_ASYNC_LDS
        a_copy64(ga, la);
#pragma unroll
        for (int q = 0; q < 16; q += 4) {
            a_copy128(wg + q, lg + q);
            a_copy128(wu + q, lu + q);
        }
#else
        *(float2*)la = *(const float2*)ga;
#pragma unroll
        for (int q = 0; q < 16; q += 4) {
            *(float4*)(lg + q) = *(const float4*)(wg + q);
            *(float4*)(lu + q) = *(const float4*)(wu + q);
        }
        if (kc + 32 < H_) {                  // global_prefetch_b8 next K-chunk
            __builtin_prefetch(wg + 32 * TWO_I, 0, 0);
            __builtin_prefetch(wu + 32 * TWO_I, 0, 0);
        }
#endif
    };

    v8f accG = {};
    v8f accU = {};

    stage(0, 0);

    for (int kc = 0; kc < H_; kc += 32) {
        const int cur = (kc >> 5) & 1;
        wait_async_zero();       // own async copies for buffer `cur` landed
        __syncthreads();         // all waves: cur ready; prev buffer free to overwrite
        if (kc + 32 < H_) stage(kc + 32, cur ^ 1);

        const float* Asb = As[cur];
        const float* Bgb = Bg[cur];
        const float* Bub = Bu[cur];
#pragma unroll
        for (int ks = 0; ks < 32; ks += 4) {
            // A frag: lane holds row m, K = ks+koff, ks+koff+1
            v2f a;
            {
                const float2 t2 = *(const float2*)&Asb[m * AS + ks + koff];
                a.x = t2.x; a.y = t2.y;
            }
            const int kr = ks + half;        // B frag: rows kr (v0) and kr+2 (v1)
            v2f b;
            b.x = Bgb[kr * BS + cb + m];
            b.y = Bgb[(kr + 2) * BS + cb + m];
            accG = __builtin_amdgcn_wmma_f32_16x16x4_f32(false, a, false, b,
                                                         (short)0, accG, false, false);
            b.x = Bub[kr * BS + cb + m];
            b.y = Bub[(kr + 2) * BS + cb + m];
            accU = __builtin_amdgcn_wmma_f32_16x16x4_f32(false, a, false, b,
                                                         (short)0, accU, false, false);
        }
    }

    // SwiGLU fused in registers; C layout: VGPR i -> M = i + half*8, N = lane&15
#pragma unroll
    for (int i = 0; i < 8; ++i) {
        const int M = i + half * 8;
        const int p = rowStart + M;
        if (p < segEnd) {
            const float g = accG[i];
            const float u = accU[i];
            const float s = g / (1.0f + __expf(-g));   // silu
            gated[(size_t)p * I_ + n0 + cb + m] = u * s;
        }
    }
}

// ---------------------------------------------------------------------------
// Kernel: down-proj GEMM + weighted scatter-add into output.
// Block = (tile of 16 pairs) x (128 columns of H). Same pipeline as k_gateup.
// ---------------------------------------------------------------------------
__global__ __launch_bounds__(256)
void k_down(const float* __restrict__ gated, const float* __restrict__ dn,
            const int* __restrict__ pair_token, const float* __restrict__ pair_w,
            const int* __restrict__ tile_e, const int* __restrict__ tile_row,
            const int* __restrict__ tile_end, const int* __restrict__ numTiles,
            float* __restrict__ out) {
    const int tile = blockIdx.x;
    if (tile >= *numTiles) return;
    const int e        = tile_e[tile];
    const int rowStart = tile_row[tile];
    const int segEnd   = tile_end[tile];
    const int n0       = blockIdx.y * 128;   // column group within H

    const int tid  = threadIdx.x;
    const int lane = tid & 31;
    const int wv   = tid >> 5;
    const int cb   = wv * 16;

    constexpr int AS = 36;
    constexpr int BS = 144;
    __shared__ float As[2][16 * AS];
    __shared__ float Bd[2][32 * BS];

    const int arow = tid >> 4;
    const int acol = (tid & 15) * 2;
    int prowA = rowStart + arow;
    if (prowA >= segEnd) prowA = segEnd - 1;
    const float* grow = gated + (size_t)prowA * I_;   // contiguous gathered rows

    const int brow = tid >> 3;
    const int bcol = (tid & 7) * 16;
    const float* wbase = dn + (size_t)e * I_ * H_;

    const int m    = lane & 15;
    const int half = lane >> 4;
    const int koff = half * 2;

    auto stage = [&](int kc, int buf) {
        const float* ga = grow + kc + acol;
        float*       la = &As[buf][arow * AS + acol];
        const float* wd = wbase + (size_t)(kc + brow) * H_ + n0 + bcol;
        float*       ld = &Bd[buf][brow * BS + bcol];
#if USE_ASYNC_LDS
        a_copy64(ga, la);
#pragma unroll
        for (int q = 0; q < 16; q += 4) a_copy128(wd + q, ld + q);
#else
        *(float2*)la = *(const float2*)ga;
#pragma unroll
        for (int q = 0; q < 16; q += 4) *(float4*)(ld + q) = *(const float4*)(wd + q);
        if (kc + 32 < I_) __builtin_prefetch(wd + 32 * H_, 0, 0);
#endif
    };

    v8f acc = {};

    stage(0, 0);

    for (int kc = 0; kc < I_; kc += 32) {
        const int cur = (kc >> 5) & 1;
        wait_async_zero();
        __syncthreads();
        if (kc + 32 < I_) stage(kc + 32, cur ^ 1);

        const float* Asb = As[cur];
        const float* Bdb = Bd[cur];
#pragma unroll
        for (int ks = 0; ks < 32; ks += 4) {
            v2f a;
            {
                const float2 t2 = *(const float2*)&Asb[m * AS + ks + koff];
                a.x = t2.x; a.y = t2.y;
            }
            const int kr = ks + half;
            v2f b;
            b.x = Bdb[kr * BS + cb + m];
            b.y = Bdb[(kr + 2) * BS + cb + m];
            acc = __builtin_amdgcn_wmma_f32_16x16x4_f32(false, a, false, b,
                                                        (short)0, acc, false, false);
        }
    }

#pragma unroll
    for (int i = 0; i < 8; ++i) {
        const int M = i + half * 8;
        const int p = rowStart + M;
        if (p < segEnd) {
            const int   tok = pair_token[p];
            const float w   = pair_w[p];
            atomicAdd(&out[(size_t)tok * H_ + n0 + cb + m], w * acc[i]);
        }
    }
}

// ---------------------------------------------------------------------------
// Launch
// ---------------------------------------------------------------------------
extern "C" void kernel_launch(void* const* d_in, const int* in_sizes, int n_in,
                              void* d_out, int out_size, void* d_ws, size_t ws_size,
                              hipStream_t stream) {
    const float* hs   = (const float*)d_in[0];   // [T, H]
    const float* rw   = (const float*)d_in[1];   // [T, K]
    const int*   ridx = (const int*)  d_in[2];   // [T, K]
    // d_in[3] router_logits: only used for shape in reference
    const float* gup  = (const float*)d_in[4];   // [E*H, 2I]
    const float* dnp  = (const float*)d_in[5];   // [E*I, H]
    float* out = (float*)d_out;                  // [T, 1, H]

    // Workspace layout (ints)
    int* wsi        = (int*)d_ws;
    int* cnt        = wsi + 0;                   // [32]
    int* off        = wsi + 32;                  // [32]
    int* cursor     = wsi + 64;                  // [32]
    int* numTiles   = wsi + 96;                  // [1]
    int* tile_e     = wsi + 128;                 // [MAX_TILES]
    int* tile_row   = wsi + 128 + MAX_TILES;     // [MAX_TILES]
    int* tile_end   = wsi + 128 + 2 * MAX_TILES; // [MAX_TILES]
    int* pair_token = wsi + 2048;                // [NPAIR]
    float* pair_w   = (float*)(wsi + 2048 + NPAIR);        // [NPAIR]
    float* gated    = (float*)(wsi + 2048 + 2 * NPAIR);    // [NPAIR, I] ~25.2 MB
    (void)in_sizes; (void)n_in; (void)out_size; (void)ws_size;

    k_route_zero<<<1, 32, 0, stream>>>(cnt);
    k_zero_out<<<(T_ * H_ / 4) / 256, 256, 0, stream>>>((float4*)out);
    k_count<<<NPAIR / 256, 256, 0, stream>>>(ridx, cnt);
    k_scan<<<1, 1, 0, stream>>>(cnt, off, cursor, tile_e, tile_row, tile_end, numTiles);
    k_scatter<<<NPAIR / 256, 256, 0, stream>>>(ridx, rw, cursor, pair_token, pair_w);

    dim3 g2(MAX_TILES, I_ / 128, 1);   // (row tiles, 6 column groups of I)
    k_gateup<<<g2, 256, 0, stream>>>(hs, gup, pair_token, tile_e, tile_row, tile_end,
                                     numTiles, gated);

    dim3 g3(MAX_TILES, H_ / 128, 1);   // (row tiles, 16 column groups of H)
    k_down<<<g3, 256, 0, stream>>>(gated, dnp, pair_token, pair_w, tile_e, tile_row,
                                   tile_end, numTiles, out);
}